// MultVariateKLD1_75840532512889
// MI455X (gfx1250) — compile-verified
//
#include <hip/hip_runtime.h>

typedef __attribute__((ext_vector_type(2))) float v2f;
typedef __attribute__((ext_vector_type(4))) float v4f;
typedef __attribute__((ext_vector_type(8))) float v8f;

// Dual cross-lane reduction of (s, q) over a wave32 using one f32 WMMA.
// A-matrix 16x4 f32 layout: lane m      -> A[m][0]=a.x, A[m][1]=a.y
//                           lane m+16   -> A[m][2]=a.x, A[m][3]=a.y
// Selector B (4x16): B[0][n]=B[2][n]=(n<8), B[1][n]=B[3][n]=(n>=8)
//  => D[m][n<8]  = s_m + s_{m+16}
//     D[m][n>=8] = q_m + q_{m+16}
// Summing the 8 accumulator VGPRs folds M=0..7 (lanes 0-15) / M=8..15
// (lanes 16-31); one xor-16 swap finishes the M reduction; xor-8 swap
// exchanges the S / Q columns so every lane ends with both totals.
__device__ __forceinline__ void wave_reduce2_wmma(float s, float q,
                                                  float& S, float& Q) {
    const int lane = (int)(threadIdx.x & 31u);
    const int n = lane & 15;
    v2f a; a.x = s; a.y = q;
    v2f b; b.x = (n < 8) ? 1.0f : 0.0f;
           b.y = (n < 8) ? 0.0f : 1.0f;
    v8f c = {};
    c = __builtin_amdgcn_wmma_f32_16x16x4_f32(
            /*neg_a=*/false, a, /*neg_b=*/false, b,
            /*c_mod=*/(short)0, c, /*reuse_a=*/false, /*reuse_b=*/false);
    float t = ((c[0] + c[1]) + (c[2] + c[3])) + ((c[4] + c[5]) + (c[6] + c[7]));
    t += __shfl_xor(t, 16, 32);          // fold lane halves (M=0..7 + M=8..15)
    float o = __shfl_xor(t, 8, 32);      // fetch the other column group
    S = (n < 8) ? t : o;
    Q = (n < 8) ? o : t;
}

__global__ __launch_bounds__(256) void kld_stats_kernel(
    const float* __restrict__ m1, const float* __restrict__ m2,
    float* __restrict__ ws, int rows) {
    constexpr int D = 512;
    const int lane = (int)(threadIdx.x & 31u);
    const int wid  = (int)(threadIdx.x >> 5);
    const int wave = (int)blockIdx.x * 8 + wid;
    const int nwaves = (int)gridDim.x * 8;

    float acc = 0.0f;  // meaningful on lane 0 of each wave

    for (int r = wave; r < rows; r += nwaves) {
        const v4f* p1 = (const v4f*)(m1 + (size_t)r * D);
        const v4f* p2 = (const v4f*)(m2 + (size_t)r * D);
        float s1 = 0.f, q1 = 0.f, s2 = 0.f, q2 = 0.f;
#pragma unroll
        for (int i = 0; i < 4; ++i) {
            // 512 floats/row = 128 float4; lanes stride contiguously:
            // each instruction touches 512 consecutive bytes. NT hint:
            // 512 MiB streamed once, don't thrash L2.
            v4f x = __builtin_nontemporal_load(&p1[lane + 32 * i]);
            v4f y = __builtin_nontemporal_load(&p2[lane + 32 * i]);
            s1 += (x.x + x.y) + (x.z + x.w);
            q1 = fmaf(x.x, x.x, fmaf(x.y, x.y, fmaf(x.z, x.z, fmaf(x.w, x.w, q1))));
            s2 += (y.x + y.y) + (y.z + y.w);
            q2 = fmaf(y.x, y.x, fmaf(y.y, y.y, fmaf(y.z, y.z, fmaf(y.w, y.w, q2))));
        }

        float S1, Q1, S2, Q2;
        wave_reduce2_wmma(s1, q1, S1, Q1);
        wave_reduce2_wmma(s2, q2, S2, Q2);

        if (lane == 0) {
            const float inv_d   = 1.0f / 512.0f;
            const float inv_dm1 = 1.0f / 511.0f;
            float sig1 = S1 * inv_d;                       // row mean m1
            float sig2 = S2 * inv_d;                       // row mean m2
            float var1 = (Q1 - S1 * S1 * inv_d) * inv_dm1; // unbiased var m1
            float var2 = (Q2 - S2 * S2 * inv_d) * inv_dm1; // unbiased var m2
            float sigA = 0.5f * (sig1 + sig2);
            float invA = 1.0f / sigA;
            float dmu  = 0.5f * (var2 - var1);             // mu_avg - mu1 = -(mu_avg - mu2)
            // t1_1 + t1_2 + t2_1 + t2_2 + t3_1 + t3_2 for this row
            acc += 2.0f * logf(sigA) - logf(sig1) - logf(sig2)
                 + (sig1 + sig2) * invA
                 + 2.0f * dmu * dmu * invA;
        }
    }

    __shared__ float slots[8];
    if (lane == 0) slots[wid] = acc;
    __syncthreads();
    if (threadIdx.x == 0) {
        float b = 0.f;
#pragma unroll
        for (int i = 0; i < 8; ++i) b += slots[i];  // fixed order: deterministic
        ws[blockIdx.x] = b;
    }
}

__global__ __launch_bounds__(256) void kld_finalize_kernel(
    const float* __restrict__ ws, float* __restrict__ out,
    int nparts, float n_len, float inv_bs) {
    __shared__ float sm[256];
    float s = 0.f;
    for (int i = (int)threadIdx.x; i < nparts; i += 256) s += ws[i];
    sm[threadIdx.x] = s;
    __syncthreads();
    for (int off = 128; off > 0; off >>= 1) {
        if ((int)threadIdx.x < off) sm[threadIdx.x] += sm[threadIdx.x + off];
        __syncthreads();
    }
    if (threadIdx.x == 0)
        out[0] = 0.25f * sm[0] * inv_bs - 0.5f * n_len;
}

extern "C" void kernel_launch(void* const* d_in, const int* in_sizes, int n_in,
                              void* d_out, int out_size, void* d_ws, size_t ws_size,
                              hipStream_t stream) {
    const float* m1 = (const float*)d_in[0];
    const float* m2 = (const float*)d_in[1];
    float* out = (float*)d_out;
    float* ws  = (float*)d_ws;

    const int D    = 512;                 // last dim per reference
    const int SLEN = 4096;                // m1.shape[1] per reference
    const int rows = in_sizes[0] / D;     // bs * slen = 131072
    const int bs   = rows / SLEN;         // 32

    const int blocks = 4096;              // 8 waves/block -> 32768 waves, 4 rows each
    kld_stats_kernel<<<blocks, 256, 0, stream>>>(m1, m2, ws, rows);
    kld_finalize_kernel<<<1, 256, 0, stream>>>(ws, out, blocks,
                                               (float)SLEN, 1.0f / (float)bs);
}